// UniterPostProcessor_16269336118082
// MI455X (gfx1250) — compile-verified
//
#include <hip/hip_runtime.h>
#include <hip/hip_bf16.h>
#include <stdint.h>

// ---------------------------------------------------------------------------
// Shapes
// ---------------------------------------------------------------------------
#define BB 64
#define RR 4096
#define NN 80
#define OBJ_C 151
#define REL_C 51
#define NPAD 160           // 151 padded to 10 WMMA n-tiles of 16 (output only)

// d_out layout (floats), concatenated in reference return order:
//   obj_pred   [B,80]        off 0
//   obj_scores [B,80]        off 5120
//   pairs[ord] [B,R,2]       off 10240
//   rel_prob   [B,R,51]      off 534528
//   rel_class  [B,R]         off 13903872
#define O_PRED  0L
#define O_OSC   5120L
#define O_PAIR  10240L
#define O_PROB  534528L
#define O_RCLS  13903872L

typedef float v2f __attribute__((ext_vector_type(2)));
typedef float v8f __attribute__((ext_vector_type(8)));
typedef unsigned int v4u __attribute__((ext_vector_type(4)));
typedef int v4i __attribute__((ext_vector_type(4)));
typedef int v8i __attribute__((ext_vector_type(8)));

__device__ __forceinline__ v8f wmma4(v2f a, v2f b, v8f c) {
  // V_WMMA_F32_16X16X4_F32: D = A(16x4) * B(4x16) + C(16x16)
  return __builtin_amdgcn_wmma_f32_16x16x4_f32(false, a, false, b, (short)0, c,
                                               false, false);
}

// ---------------------------------------------------------------------------
// TDM: DMA one 32x151 f32 tile from global memory into LDS (ISA ch.8 D#).
// Group0: count=1 | lds_addr | global_addr[56:0] | type=2
// Group1: data_size=4B | tensor_dim0=151 | tensor_dim1=32 | tile_dim0=151 |
//         tile_dim1=32 | tensor_dim0_stride=151   (2-D tensor: groups 2/3 = 0)
// ---------------------------------------------------------------------------
#define RCHUNK 32
#define LROW OBJ_C          // TDM writes contiguous rows: LDS row stride = 151

__device__ __forceinline__ void tdm_load_tile(const float* gptr,
                                              const float* lptr) {
  const unsigned long long ga = (unsigned long long)(uintptr_t)gptr;
  const unsigned galo = __builtin_amdgcn_readfirstlane((unsigned)ga);
  const unsigned gahi = __builtin_amdgcn_readfirstlane((unsigned)(ga >> 32));
  const unsigned lds =
      __builtin_amdgcn_readfirstlane((unsigned)(uintptr_t)lptr);
  v4u g0;
  g0[0] = 1u;                                  // count=1 (user mode)
  g0[1] = lds;                                 // lds_addr
  g0[2] = galo;                                // global_addr[31:0]
  g0[3] = (gahi & 0x1ffffffu) | (2u << 30);    // global_addr[56:32] | type=2
  v8i g1;
  g1[0] = (int)(2u << 16);                     // data_size = 4 bytes
  g1[1] = (int)((unsigned)OBJ_C << 16);        // tensor_dim0[15:0]  @bit48
  g1[2] = (int)((unsigned)RCHUNK << 16);       // tensor_dim1[15:0]  @bit80
  g1[3] = (int)((unsigned)OBJ_C << 16);        // tile_dim0          @bit112
  g1[4] = (int)(unsigned)RCHUNK;               // tile_dim1          @bit128
  g1[5] = (int)(unsigned)OBJ_C;                // tensor_dim0_stride[31:0]
  g1[6] = 0;
  g1[7] = 0;
  const v4i z4 = {0, 0, 0, 0};
  const v8i z8 = {0, 0, 0, 0, 0, 0, 0, 0};
  // 6-arg toolchain form: (g0, g1, g2, g3, g4, cpol)
  __builtin_amdgcn_tensor_load_to_lds(g0, g1, z4, z4, z8, 0);
}

// ---------------------------------------------------------------------------
// Kernel 1: segment sums as WMMA matmul with TDM double-buffered staging.
//   sums[b][80][160] = onehot(sub)^T * sub_logits + onehot(obj)^T * obj_logits
// One block (25 waves = 800 threads) per image. Wave 0 issues TDM loads one
// chunk ahead; each wave owns 2 output tiles (mtile = w/5, ntiles w%5, w%5+5).
// ---------------------------------------------------------------------------
__global__ void __launch_bounds__(800)
seg_sums_wmma(const float* __restrict__ sub_logits,
              const float* __restrict__ obj_logits,
              const int* __restrict__ pairs,
              float* __restrict__ sums) {
  __shared__ float sSub[2][RCHUNK * LROW];
  __shared__ float sObj[2][RCHUNK * LROW];
  __shared__ int   sIS[2][RCHUNK];
  __shared__ int   sIO[2][RCHUNK];

  const int b    = blockIdx.x;
  const int tid  = threadIdx.x;
  const int wave = tid >> 5;        // 0..24
  const int lane = tid & 31;
  const int h    = lane >> 4;       // lane half (K group for A/B fragments)
  const int n    = lane & 15;       // A: row M%16 ; B/C: col N%16
  const int mtile = wave / 5;
  const int p     = wave % 5;
  const int mbase = mtile * 16;
  const int nb0   = p * 16;
  const int nb1   = nb0 + 80;

  v8f c0 = {};
  v8f c1 = {};

  const long base = (long)b * RR * OBJ_C;
  const int NCH = RR / RCHUNK;      // 128 chunks

  // Prologue: stage chunk 0 (TDM tiles + indices) into buffer 0.
  if (tid < 32) {
    tdm_load_tile(sub_logits + base, &sSub[0][0]);
    tdm_load_tile(obj_logits + base, &sObj[0][0]);
    sIS[0][lane] = pairs[((long)b * RR + lane) * 2 + 0];
    sIO[0][lane] = pairs[((long)b * RR + lane) * 2 + 1];
  }

  for (int i = 0; i < NCH; ++i) {
    const int cur = i & 1;
    if (tid < 32) {
      if (i + 1 < NCH) {
        const long r1 = (long)(i + 1) * RCHUNK;
        tdm_load_tile(sub_logits + base + r1 * OBJ_C, &sSub[cur ^ 1][0]);
        tdm_load_tile(obj_logits + base + r1 * OBJ_C, &sObj[cur ^ 1][0]);
        sIS[cur ^ 1][lane] = pairs[((long)b * RR + r1 + lane) * 2 + 0];
        sIO[cur ^ 1][lane] = pairs[((long)b * RR + r1 + lane) * 2 + 1];
        __builtin_amdgcn_s_wait_tensorcnt(2);  // chunk i's 2 tiles landed
      } else {
        __builtin_amdgcn_s_wait_tensorcnt(0);
      }
    }
    __syncthreads();   // publish chunk i (TDM data + indices) to all waves

    const float* Sb = &sSub[cur][0];
    const float* Ob = &sObj[cur][0];
    const int*   IS = &sIS[cur][0];
    const int*   IO = &sIO[cur][0];

#pragma unroll
    for (int k4 = 0; k4 < RCHUNK; k4 += 4) {
      const int k = k4 + 2 * h;              // K index for this lane-half
      const int is0 = IS[k], is1 = IS[k + 1];
      const int io0 = IO[k], io1 = IO[k + 1];
      v2f aS, aO, bS0, bS1, bO0, bO1;
      // One-hot A fragments (A 16x4: lane%16 = M, {vgpr,half} = K)
      aS.x = (is0 == mbase + n) ? 1.f : 0.f;
      aS.y = (is1 == mbase + n) ? 1.f : 0.f;
      aO.x = (io0 == mbase + n) ? 1.f : 0.f;
      aO.y = (io1 == mbase + n) ? 1.f : 0.f;
      // B fragments from LDS (B 4x16: lane%16 = N, {vgpr,half} = K)
      bS0.x = Sb[k * LROW + nb0 + n];
      bS0.y = Sb[(k + 1) * LROW + nb0 + n];
      bS1.x = Sb[k * LROW + nb1 + n];
      bS1.y = Sb[(k + 1) * LROW + nb1 + n];
      bO0.x = Ob[k * LROW + nb0 + n];
      bO0.y = Ob[(k + 1) * LROW + nb0 + n];
      bO1.x = Ob[k * LROW + nb1 + n];
      bO1.y = Ob[(k + 1) * LROW + nb1 + n];
      c0 = wmma4(aS, bS0, c0);
      c1 = wmma4(aS, bS1, c1);
      c0 = wmma4(aO, bO0, c0);
      c1 = wmma4(aO, bO1, c1);
    }
    __syncthreads();   // all waves done reading buf[cur] before it is reused
  }

  // C/D layout: vgpr v, lane-half h -> M = v + 8h ; lane%16 -> N
  // (cols >=151 contain cross-row garbage from the unpadded LDS tiles; they
  //  land in sums[:,151:160] which no consumer reads.)
  float* dst = sums + (long)b * NN * NPAD;
#pragma unroll
  for (int v = 0; v < 8; ++v) {
    const int m = mbase + v + 8 * h;
    dst[m * NPAD + nb0 + n] = c0[v];
    dst[m * NPAD + nb1 + n] = c1[v];
  }
}

// ---------------------------------------------------------------------------
// Kernel 2: counts (LDS float atomics) + refine + softmax + obj score/argmax.
// One block (256 thr = 8 waves) per image; one wave per segment row.
// ---------------------------------------------------------------------------
__global__ void __launch_bounds__(256)
refine_softmax(const float* __restrict__ sums,
               const int* __restrict__ pairs,
               float* __restrict__ out,
               float* __restrict__ objsc) {
  __shared__ float cnt[NN];
  __shared__ float buf[8][152];
  const int b = blockIdx.x, tid = threadIdx.x;
  if (tid < NN) cnt[tid] = 0.f;
  __syncthreads();
  for (int r = tid; r < RR; r += 256) {
    const int s = pairs[((long)b * RR + r) * 2 + 0];
    const int o = pairs[((long)b * RR + r) * 2 + 1];
    atomicAdd(&cnt[s], 1.f);   // ds_add_f32
    atomicAdd(&cnt[o], 1.f);
  }
  __syncthreads();

  const int wave = tid >> 5, lane = tid & 31;
  for (int m = wave; m < NN; m += 8) {
    const float inv = 1.f / fmaxf(cnt[m], 1.f);
    const float* row = sums + (long)b * NN * NPAD + m * NPAD;
    float mx = -3.4e38f;
    for (int c = lane; c < OBJ_C; c += 32) {
      const float v = row[c] * inv;       // refine
      buf[wave][c] = v;
      mx = fmaxf(mx, v);
    }
#pragma unroll
    for (int off = 16; off; off >>= 1) mx = fmaxf(mx, __shfl_xor(mx, off, 32));
    float sum = 0.f, bv = -3.4e38f;
    int bi = OBJ_C;
    for (int c = lane; c < OBJ_C; c += 32) {
      const float v = buf[wave][c];
      sum += __expf(v - mx);              // softmax over ALL 151 cols
      if (c >= 1 && v > bv) { bv = v; bi = c; }  // argmax over cols 1..150
    }
#pragma unroll
    for (int off = 16; off; off >>= 1) sum += __shfl_xor(sum, off, 32);
#pragma unroll
    for (int off = 16; off; off >>= 1) {
      const float ov = __shfl_xor(bv, off, 32);
      const int   oi = __shfl_xor(bi, off, 32);
      if (ov > bv || (ov == bv && oi < bi)) { bv = ov; bi = oi; }
    }
    if (lane == 0) {
      const float sc = __expf(bv - mx) / sum;
      out[O_PRED + (long)b * NN + m] = (float)bi;  // class index (already +1'd)
      out[O_OSC  + (long)b * NN + m] = sc;
      objsc[b * NN + m] = sc;
    }
  }
}

// ---------------------------------------------------------------------------
// Kernel 3: per-relation softmax stats + stable 64-bit sort key.
// key = (~orderable(triple) << 32) | index  -> ascending u64 sort ==
//       stable descending sort on triple (matches jnp stable argsort(-t)).
// ---------------------------------------------------------------------------
__global__ void __launch_bounds__(256)
rel_softmax_keys(const float* __restrict__ rel_logits,
                 const int* __restrict__ pairs,
                 const float* __restrict__ objsc,
                 float* __restrict__ m_ws, float* __restrict__ s_ws,
                 int* __restrict__ cls_ws,
                 unsigned long long* __restrict__ key_ws) {
  const long r = (long)blockIdx.x * 256 + threadIdx.x;
  if (r >= (long)BB * RR) return;
  const int b = (int)(r >> 12);
  const int j = (int)(r & (RR - 1));
  const float* x = rel_logits + r * REL_C;
  float v[REL_C];
  float mx = -3.4e38f, bv = -3.4e38f;
  int bi = 1;
#pragma unroll
  for (int c = 0; c < REL_C; ++c) {
    v[c] = x[c];
    mx = fmaxf(mx, v[c]);
    if (c >= 1 && v[c] > bv) { bv = v[c]; bi = c; }
  }
  float sum = 0.f;
#pragma unroll
  for (int c = 0; c < REL_C; ++c) sum += __expf(v[c] - mx);
  const float relsc = __expf(bv - mx) / sum;
  const int si = pairs[r * 2 + 0], oi = pairs[r * 2 + 1];
  const float triple = relsc * objsc[b * NN + si] * objsc[b * NN + oi];
  m_ws[r] = mx;
  s_ws[r] = sum;
  cls_ws[r] = bi;
  unsigned u = __float_as_uint(triple);
  u = (u & 0x80000000u) ? ~u : (u | 0x80000000u);  // ascending-orderable
  key_ws[r] = ((unsigned long long)(~u) << 32) | (unsigned)j;
}

// ---------------------------------------------------------------------------
// Kernel 4: in-LDS bitonic sort of 4096 u64 keys per image (ascending).
// ---------------------------------------------------------------------------
__global__ void __launch_bounds__(1024)
sort_keys(unsigned long long* __restrict__ key_ws, int* __restrict__ perm) {
  __shared__ unsigned long long k[RR];   // 32 KB
  const int b = blockIdx.x, tid = threadIdx.x;
  for (int t = tid; t < RR; t += 1024) k[t] = key_ws[(long)b * RR + t];
  for (unsigned size = 2; size <= RR; size <<= 1) {
    for (unsigned stride = size >> 1; stride > 0; stride >>= 1) {
      __syncthreads();
      for (unsigned t = tid; t < RR / 2; t += 1024) {
        const unsigned i = 2 * t - (t & (stride - 1));
        const unsigned jx = i + stride;
        const bool up = ((i & size) == 0);
        const unsigned long long a = k[i], c = k[jx];
        if ((a > c) == up) { k[i] = c; k[jx] = a; }
      }
    }
  }
  __syncthreads();
  for (int t = tid; t < RR; t += 1024)
    perm[(long)b * RR + t] = (int)(k[t] & 0xffffffffu);
}

// ---------------------------------------------------------------------------
// Kernel 5: gather/emit. 64 threads per output slot; lanes 0..50 recompute
// rel_prob[order] from logits + stored (max,sum) — coalesced in class dim.
// ---------------------------------------------------------------------------
__global__ void __launch_bounds__(256)
gather_emit(const float* __restrict__ rel_logits,
            const int* __restrict__ pairs,
            const int* __restrict__ perm,
            const float* __restrict__ m_ws, const float* __restrict__ s_ws,
            const int* __restrict__ cls_ws,
            float* __restrict__ out) {
  const int g = threadIdx.x >> 6;                // slot group 0..3
  const int l = threadIdx.x & 63;
  const long j = (long)blockIdx.x * 4 + g;       // output slot over B*R
  if (j >= (long)BB * RR) return;
  const int b = (int)(j >> 12);
  const int i = perm[j];
  const long src = ((long)b << 12) + i;
  if (l < REL_C) {
    const float e = __expf(rel_logits[src * REL_C + l] - m_ws[src]) / s_ws[src];
    out[O_PROB + j * REL_C + l] = e;
  } else if (l == REL_C) {
    out[O_PAIR + j * 2 + 0] = (float)pairs[src * 2 + 0];
  } else if (l == REL_C + 1) {
    out[O_PAIR + j * 2 + 1] = (float)pairs[src * 2 + 1];
  } else if (l == REL_C + 2) {
    out[O_RCLS + j] = (float)cls_ws[src];
  }
}

// ---------------------------------------------------------------------------
extern "C" void kernel_launch(void* const* d_in, const int* in_sizes, int n_in,
                              void* d_out, int out_size, void* d_ws,
                              size_t ws_size, hipStream_t stream) {
  const float* rel_logits = (const float*)d_in[0];
  const float* sub_logits = (const float*)d_in[1];
  const float* obj_logits = (const float*)d_in[2];
  const int*   pairs      = (const int*)d_in[3];
  float* out = (float*)d_out;

  char* w = (char*)d_ws;
  float* sums = (float*)w;                       w += (size_t)BB * NN * NPAD * 4;
  float* objsc = (float*)w;                      w += (size_t)BB * NN * 4;
  float* m_ws = (float*)w;                       w += (size_t)BB * RR * 4;
  float* s_ws = (float*)w;                       w += (size_t)BB * RR * 4;
  int*   cls_ws = (int*)w;                       w += (size_t)BB * RR * 4;
  w = (char*)(((uintptr_t)w + 7) & ~(uintptr_t)7);
  unsigned long long* key_ws = (unsigned long long*)w;
  w += (size_t)BB * RR * 8;
  int* perm = (int*)w;

  seg_sums_wmma<<<BB, 800, 0, stream>>>(sub_logits, obj_logits, pairs, sums);
  refine_softmax<<<BB, 256, 0, stream>>>(sums, pairs, out, objsc);
  rel_softmax_keys<<<(BB * RR) / 256, 256, 0, stream>>>(
      rel_logits, pairs, objsc, m_ws, s_ws, cls_ws, key_ws);
  sort_keys<<<BB, 1024, 0, stream>>>(key_ws, perm);
  gather_emit<<<(BB * RR) / 4, 256, 0, stream>>>(rel_logits, pairs, perm, m_ws,
                                                 s_ws, cls_ws, out);
}